// YOLO_loss_86655260164796
// MI455X (gfx1250) — compile-verified
//
#include <hip/hip_runtime.h>

// ---- problem dims (from reference) ----
#define BB 128
#define CC 5
#define HH 128
#define WW 128
#define HWP (HH * WW)          // 16384
#define BHW (BB * HWP)         // 2,097,152 cells
#define NGROUP (BHW / 4)       // 524,288 float4 groups (HWP % 4 == 0 -> never crosses b)
#define NBLOCKS 1024           // partials produced by kernel 1 (= 256 float4s)

typedef __attribute__((ext_vector_type(4))) float f32x4;
typedef __attribute__((ext_vector_type(2))) float v2f;
typedef __attribute__((ext_vector_type(8))) float v8f;

// non-temporal 128-bit load: data is streamed exactly once -> TH=NT,
// don't let the 168MB stream churn the 192MB L2.
__device__ __forceinline__ f32x4 ntld(const float* p) {
  return __builtin_nontemporal_load(reinterpret_cast<const f32x4*>(p));
}

// Kernel 1: streaming masked SSE, one f32 partial per block.
// Roofline: 167.8 MB read once / 23.3 TB/s ~= 7.2 us; compute is negligible,
// so the only job is full-line coalesced b128 streaming.
__global__ void yolo_partial(const float* __restrict__ la,
                             const float* __restrict__ lb,
                             const float* __restrict__ pab,
                             const float* __restrict__ pba,
                             float* __restrict__ ws) {
  float acc = 0.0f;
  const int stride = gridDim.x * blockDim.x;
  for (int g = blockIdx.x * blockDim.x + threadIdx.x; g < NGROUP; g += stride) {
    const int cell = g << 2;               // 4 consecutive cells, same b
    const int b    = cell / HWP;
    const int hw   = cell - b * HWP;
    const int base = b * (CC * HWP) + hw;  // [B,C,H,W] flat: b*C*HW + c*HW + hw

    const f32x4 m = ntld(la + base);       // labela channel 0 = mask (and SSE term)
    f32x4 cs = {0.0f, 0.0f, 0.0f, 0.0f};
#pragma unroll
    for (int c = 0; c < CC; ++c) {
      const int off = base + c * HWP;
      const f32x4 a = (c == 0) ? m : ntld(la + off);
      const f32x4 p = ntld(pab + off);
      const f32x4 l = ntld(lb  + off);
      const f32x4 q = ntld(pba + off);
      const f32x4 d0 = a - p;
      const f32x4 d1 = l - q;
      cs += d0 * d0 + d1 * d1;
    }
    // branchless mask gate (10% density: skipping loads would still touch ~97% of lines)
    acc += (m.x != 0.0f ? cs.x : 0.0f);
    acc += (m.y != 0.0f ? cs.y : 0.0f);
    acc += (m.z != 0.0f ? cs.z : 0.0f);
    acc += (m.w != 0.0f ? cs.w : 0.0f);
  }

  // wave32 reduction
#pragma unroll
  for (int off = 16; off > 0; off >>= 1) acc += __shfl_xor(acc, off, 32);

  __shared__ float smem[8];                // 256 threads = 8 waves (wave32)
  const int lane = threadIdx.x & 31;
  const int wid  = threadIdx.x >> 5;
  if (lane == 0) smem[wid] = acc;
  __syncthreads();
  if (wid == 0) {
    float v = (lane < (int)(blockDim.x >> 5)) ? smem[lane] : 0.0f;
#pragma unroll
    for (int off = 4; off > 0; off >>= 1) v += __shfl_xor(v, off, 32);
    if (lane == 0) ws[blockIdx.x] = v;
  }
}

// Kernel 2: one wave reduces the 1024 block partials.
// 1024 floats = 256 f32x4; lane l loads vectors l, l+32, ..., l+224:
// 8 INDEPENDENT b128 loads issued before any wait -> one L2 round trip
// (the previous scalar loop serialized 32 dependent loads = ~10 us of latency).
// Cross-lane sum is then one v_wmma_f32_16x16x4_f32:
//   A(16x4): VGPR0 = per-lane partials (A[m][0]=s_m, A[m][2]=s_{m+16}), VGPR1 = 0
//   B(4x16): all ones  =>  D[m][n] = s_m + s_{m+16}  (exact f32, products are x1.0)
// C/D layout puts t_0..t_7 in lanes 0-15 regs 0..7 and t_8..t_15 in lanes 16-31,
// so 7 in-lane adds + one shfl_xor(16) finish the 32-lane reduction.
__global__ void yolo_final(const float* __restrict__ ws, float* __restrict__ out) {
  const f32x4* w4 = reinterpret_cast<const f32x4*>(ws);
  const int l = threadIdx.x;
  const f32x4 v0 = w4[l + 0 * 32];
  const f32x4 v1 = w4[l + 1 * 32];
  const f32x4 v2 = w4[l + 2 * 32];
  const f32x4 v3 = w4[l + 3 * 32];
  const f32x4 v4 = w4[l + 4 * 32];
  const f32x4 v5 = w4[l + 5 * 32];
  const f32x4 v6 = w4[l + 6 * 32];
  const f32x4 v7 = w4[l + 7 * 32];
  const f32x4 t = ((v0 + v1) + (v2 + v3)) + ((v4 + v5) + (v6 + v7));
  const float s = (t.x + t.y) + (t.z + t.w);

  v2f a; a.x = s;    a.y = 0.0f;
  v2f b; b.x = 1.0f; b.y = 1.0f;
  v8f c = {};
  c = __builtin_amdgcn_wmma_f32_16x16x4_f32(
      /*neg_a=*/false, a, /*neg_b=*/false, b,
      /*c_mod=*/(short)0, c, /*reuse_a=*/false, /*reuse_b=*/false);

  float r = c[0] + c[1] + c[2] + c[3] + c[4] + c[5] + c[6] + c[7];
  float tot = r + __shfl_xor(r, 16, 32);
  if (threadIdx.x == 0) out[0] = tot;
}

extern "C" void kernel_launch(void* const* d_in, const int* in_sizes, int n_in,
                              void* d_out, int out_size, void* d_ws, size_t ws_size,
                              hipStream_t stream) {
  const float* labela  = (const float*)d_in[0];
  const float* labelb  = (const float*)d_in[1];
  const float* pred_ab = (const float*)d_in[2];
  const float* pred_ba = (const float*)d_in[3];
  float* ws  = (float*)d_ws;   // NBLOCKS f32 partials, fully rewritten every call
  float* out = (float*)d_out;  // 1 f32

  const int threads = 256;     // 8 waves/block under wave32 -> plenty of inflight loads
  yolo_partial<<<NBLOCKS, threads, 0, stream>>>(labela, labelb, pred_ab, pred_ba, ws);
  yolo_final<<<1, 32, 0, stream>>>(ws, out);
}